// DualPathRNN_69818988364145
// MI455X (gfx1250) — compile-verified
//
#include <hip/hip_runtime.h>
#include <hip/hip_bf16.h>
#include <math.h>

// ---------------- problem constants (from reference) ----------------
#define BB   4
#define CC   64
#define TT   256
#define FF   129
#define KK   8
#define HG   32
#define NN   (BB * FF)        // 516 sequences
#define LL   (TT - KK + 1)    // 249 time steps after unfold
#define M1   (LL * NN)        // 128484 GEMM-1 rows
#define KPG  (32 * KK)        // 256: per-group K (32 channels x 8 taps)
#define UCOL 512              // [g1f|g1b|g2f|g2b] each 128 (xt,f,r,xres x HG)
#define M2   (NN * TT)        // 132096 conv-GEMM rows
#define K2   (128 * KK)       // 1024 conv-GEMM K

typedef float v2f __attribute__((ext_vector_type(2)));
typedef float v8f __attribute__((ext_vector_type(8)));
typedef unsigned int u32x4 __attribute__((ext_vector_type(4)));
typedef int i32x8 __attribute__((ext_vector_type(8)));
typedef int i32x4 __attribute__((ext_vector_type(4)));

#if defined(__has_builtin)
#if __has_builtin(__builtin_amdgcn_tensor_load_to_lds) && \
    __has_builtin(__builtin_amdgcn_s_wait_tensorcnt)
#define HAVE_TDM 1
#endif
#endif
#ifndef HAVE_TDM
#define HAVE_TDM 0
#endif

// ---------------- workspace layout (floats) ----------------
// xr : (n, c, t)            NN*CC*TT           =  8,454,144 f
// U  : (l*NN+n, 512)        M1*UCOL            = 65,783,808 f
// h  : (n, ch=128, l)       NN*128*LL          = 16,445,952 f
#define WS_XR 0
#define WS_U  (NN * CC * TT)
#define WS_H  (WS_U + (size_t)M1 * UCOL)

// =====================================================================
// TDM 2D load: copy tile (tile_d0 x tile_d1) of 4-byte elements from a
// row-major tensor (row length tensor_d0, row stride stride0 elements)
// into LDS (packed, row-major).  D# layout per CDNA5 ISA ch.8.
// This toolchain exposes the 6-arg builtin:
//   (u32x4 g0, i32x8 g1, i32x4 g2, i32x4 g3, i32x8 g4, i32 cpol)
// Our tensors are 2D, so the trailing groups are zero.
// =====================================================================
#if HAVE_TDM
__device__ __forceinline__ void tdm_load_2d_f32(unsigned lds_byte_off,
                                                const void* gptr,
                                                unsigned tensor_d0, unsigned tensor_d1,
                                                unsigned tile_d0, unsigned tile_d1,
                                                unsigned stride0_elems) {
  unsigned long long ga = (unsigned long long)(uintptr_t)gptr;
  u32x4 g0;
  g0[0] = 1u;                                   // count=1, user descriptor
  g0[1] = lds_byte_off;                         // LDS byte address
  g0[2] = (unsigned)(ga & 0xFFFFFFFFu);         // global_addr[31:0]
  g0[3] = (unsigned)((ga >> 32) & 0x01FFFFFFu)  // global_addr[56:32]
        | (2u << 30);                           // type=2 ("image")
  i32x8 g1;
  g1[0] = (int)(2u << 16);                      // data_size=2 -> 4B elements
  g1[1] = (int)((tensor_d0 & 0xFFFFu) << 16);   // tensor_dim0[15:0] @bits63:48
  g1[2] = (int)(((tensor_d0 >> 16) & 0xFFFFu) | ((tensor_d1 & 0xFFFFu) << 16));
  g1[3] = (int)(((tensor_d1 >> 16) & 0xFFFFu) | ((tile_d0 & 0xFFFFu) << 16));
  g1[4] = (int)(tile_d1 & 0xFFFFu);             // tile_dim1 (tile_dim2=0)
  g1[5] = (int)stride0_elems;                   // tensor_dim0_stride[31:0]
  g1[6] = 0;
  g1[7] = 0;
  i32x4 z4 = {0, 0, 0, 0};
  i32x8 z8 = {0, 0, 0, 0, 0, 0, 0, 0};
  __builtin_amdgcn_tensor_load_to_lds(g0, g1, z4, z4, z8, 0);
}
#endif

// =====================================================================
// Kernel 1: channel LayerNorm, write xr[(b*FF+f), c, t]
// =====================================================================
__global__ void k_layernorm(const float* __restrict__ x,
                            const float* __restrict__ gamma,
                            const float* __restrict__ beta,
                            float* __restrict__ xr) {
  int idx = blockIdx.x * blockDim.x + threadIdx.x;   // over B*T*F
  if (idx >= BB * TT * FF) return;
  int f = idx % FF;
  int t = (idx / FF) % TT;
  int b = idx / (FF * TT);

  const float* xb = x + ((size_t)b * CC * TT + t) * FF + f;  // stride TT*FF per c
  float s = 0.f, ss = 0.f;
#pragma unroll 8
  for (int c = 0; c < CC; ++c) {
    float v = xb[(size_t)c * TT * FF];
    s += v; ss += v * v;
  }
  float mu   = s * (1.0f / CC);
  float var  = ss * (1.0f / CC) - mu * mu;
  float rstd = rsqrtf(var + 1e-5f);

  int n = b * FF + f;
  float* o = xr + ((size_t)n * CC) * TT + t;                 // stride TT per c
#pragma unroll 8
  for (int c = 0; c < CC; ++c) {
    float v = xb[(size_t)c * TT * FF];
    o[(size_t)c * TT] = gamma[c] * (v - mu) * rstd + beta[c];
  }
}

// =====================================================================
// Kernel 2: U = im2col(xr) @ [Wf|Wb] per group, fp32 WMMA 16x16x4.
//   block = 8 waves, each wave owns 16(M)x64(N) (4 accumulators, A reused
//   4x per load).  Weight panel 256x64 staged to LDS via TDM, shared by
//   all 8 M-tiles of the block.
//   grid = (ceil(M1/16/8), 4 col-slabs of 64, 2 groups)
// =====================================================================
__global__ void k_sru_gemm(const float* __restrict__ xr,
                           const float* __restrict__ W1f, const float* __restrict__ W1b,
                           const float* __restrict__ W2f, const float* __restrict__ W2b,
                           float* __restrict__ U) {
  extern __shared__ float ldsW[];               // [256][64] = 64KB
  const int lane = threadIdx.x & 31;
  const int wv   = threadIdx.x >> 5;            // 0..7
  const int g    = blockIdx.z;
  const int colBase = blockIdx.y * 64;          // 0,64,128,192 (within group)

  const float* Wf = g ? W2f : W1f;
  const float* Wb = g ? W2b : W1b;
  const float* Wsrc = (colBase >= 128) ? (Wb + (colBase - 128)) : (Wf + colBase);

  // ---- stage 256x64 weight panel into LDS (rows of length 64) ----
#if HAVE_TDM
  if (wv == 0) {
    tdm_load_2d_f32((unsigned)(uintptr_t)ldsW, Wsrc,
                    /*tensor_d0=*/128, /*tensor_d1=*/KPG,
                    /*tile_d0=*/64,    /*tile_d1=*/KPG,
                    /*stride0=*/128);
    __builtin_amdgcn_s_wait_tensorcnt(0);
  }
#else
  for (int e = threadIdx.x; e < KPG * 64; e += 256)
    ldsW[e] = Wsrc[(size_t)(e >> 6) * 128 + (e & 63)];
#endif
  __syncthreads();

  // ---- per-lane A coordinates (branch-free tail masking) ----
  const int tileM = blockIdx.x * 8 + wv;
  const int rowA  = tileM * 16 + (lane & 15);
  const int rowC  = rowA < M1 ? rowA : (M1 - 1);
  const float am  = rowA < M1 ? 1.0f : 0.0f;
  const int l = rowC / NN;
  const int n = rowC % NN;
  const float* aBase = xr + ((size_t)(n * CC + g * 32)) * TT + l;  // +c*TT+tap
  const int kHi  = (lane >> 4) << 1;            // lanes 16-31: K = k0+2,k0+3
  const int colL = lane & 15;

  v8f a0 = {}, a1 = {}, a2 = {}, a3 = {};
#pragma unroll 4
  for (int k0 = 0; k0 < KPG; k0 += 4) {
    const int kk0 = k0 + kHi;                   // even -> tap0,tap0+1 same chan
    const int c0 = kk0 >> 3, t0 = kk0 & 7;
    v2f a;
    a.x = aBase[(size_t)c0 * TT + t0]     * am;
    a.y = aBase[(size_t)c0 * TT + t0 + 1] * am;
    const float* bp = ldsW + kk0 * 64 + colL;   // row kk0 / kk0+1 at +64
    v2f b0; b0.x = bp[0];  b0.y = bp[64];
    v2f b1; b1.x = bp[16]; b1.y = bp[80];
    v2f b2; b2.x = bp[32]; b2.y = bp[96];
    v2f b3; b3.x = bp[48]; b3.y = bp[112];
    a0 = __builtin_amdgcn_wmma_f32_16x16x4_f32(false, a, false, b0, (short)0, a0, false, false);
    a1 = __builtin_amdgcn_wmma_f32_16x16x4_f32(false, a, false, b1, (short)0, a1, false, false);
    a2 = __builtin_amdgcn_wmma_f32_16x16x4_f32(false, a, false, b2, (short)0, a2, false, false);
    a3 = __builtin_amdgcn_wmma_f32_16x16x4_f32(false, a, false, b3, (short)0, a3, false, false);
  }

  // ---- store 16x64 tile (tail rows predicated) ----
  const int rOff = (lane >> 4) << 3;
  const size_t colU = (size_t)g * 256 + colBase + colL;
#pragma unroll
  for (int v = 0; v < 8; ++v) {
    int r = tileM * 16 + v + rOff;
    if (r < M1) {
      float* up = U + (size_t)r * UCOL + colU;
      up[0]  = a0[v];
      up[16] = a1[v];
      up[32] = a2[v];
      up[48] = a3[v];
    }
  }
}

// =====================================================================
// Kernel 3: SRU scans. thread = (dir 0..3, n, hgate). dirs: g1f,g1b,g2f,g2b
//   writes h[n, ch=dir*32+hg, l]
// =====================================================================
__global__ void k_sru_scan(const float* __restrict__ U,
                           const float* __restrict__ b1f, const float* __restrict__ b1b,
                           const float* __restrict__ b2f, const float* __restrict__ b2b,
                           float* __restrict__ H) {
  int idx = blockIdx.x * blockDim.x + threadIdx.x;
  if (idx >= 4 * NN * HG) return;
  int hg = idx & (HG - 1);
  int n  = (idx >> 5) % NN;
  int d  = idx / (NN * HG);                 // 0..3

  const float* bsel = (d == 0) ? b1f : (d == 1) ? b1b : (d == 2) ? b2f : b2b;
  const float bf = bsel[hg];
  const float br = bsel[HG + hg];
  const bool fwd = (d & 1) == 0;
  const int dbase = d * 128;

  float* hout = H + ((size_t)(n * 128 + d * 32 + hg)) * LL;
  float c = 0.0f;
  for (int s = 0; s < LL; ++s) {
    int l = fwd ? s : (LL - 1 - s);
    const float* u = U + ((size_t)(l * NN + n)) * UCOL + dbase + hg;
    float xt   = u[0];
    float fp   = u[32];
    float rp   = u[64];
    float xres = u[96];
    float fgate = 1.0f / (1.0f + __expf(-(fp + bf)));
    float rgate = 1.0f / (1.0f + __expf(-(rp + br)));
    c = fgate * c + (1.0f - fgate) * xt;
    hout[l] = rgate * c + (1.0f - rgate) * xres;
  }
}

// =====================================================================
// Kernel 4: transposed conv as WMMA GEMM + bias + residual + NCTF store.
//   out[n,o,t] = sum_{i,k, 0<=t-k<LL} h[n,i,t-k] * Wct[i,o,k]
//   block = 8 waves x (16M x 64N tile each); Wct streamed through a 64KB
//   LDS slab in 4 TDM stages (32 i-channels / 256 kk per stage).
// =====================================================================
__global__ void k_convt(const float* __restrict__ H,
                        const float* __restrict__ Wct,
                        const float* __restrict__ bct,
                        const float* __restrict__ x,
                        float* __restrict__ out) {
  extern __shared__ float ldsW[];               // 32*512 floats = 64KB slab
  const int lane = threadIdx.x & 31;
  const int wv   = threadIdx.x >> 5;
  const int tileM = blockIdx.x * 8 + wv;
  const int rowA  = tileM * 16 + (lane & 15);   // always < M2 (M2 % 128 == 0)
  const int n = rowA / TT;
  const int t = rowA % TT;
  const float* hRow = H + (size_t)n * 128 * LL; // + i*LL + l
  const int kHi  = (lane >> 4) << 1;
  const int colL = lane & 15;

  v8f a0 = {}, a1 = {}, a2 = {}, a3 = {};
  for (int s = 0; s < 4; ++s) {                 // 4 K-slabs of 256
    __syncthreads();                            // previous slab fully consumed
#if HAVE_TDM
    if (wv == 0) {
      tdm_load_2d_f32((unsigned)(uintptr_t)ldsW, Wct + (size_t)s * 32 * 512,
                      /*tensor_d0=*/512, /*tensor_d1=*/32,
                      /*tile_d0=*/512,   /*tile_d1=*/32,
                      /*stride0=*/512);
      __builtin_amdgcn_s_wait_tensorcnt(0);
    }
#else
    for (int e = threadIdx.x; e < 32 * 512; e += 256)
      ldsW[e] = Wct[(size_t)s * 32 * 512 + e];
#endif
    __syncthreads();

#pragma unroll 4
    for (int k0 = 0; k0 < 256; k0 += 4) {
      const int kk0 = s * 256 + k0 + kHi;       // even -> q0,q0+1 same i
      const int i0 = kk0 >> 3, q0 = kk0 & 7;
      // branch-free boundary masking: clamp address, zero value
      const int l0 = t - q0, l1 = l0 - 1;
      int l0c = l0 < 0 ? 0 : (l0 > LL - 1 ? LL - 1 : l0);
      int l1c = l1 < 0 ? 0 : (l1 > LL - 1 ? LL - 1 : l1);
      const float m0 = (l0 >= 0 && l0 < LL) ? 1.0f : 0.0f;
      const float m1 = (l1 >= 0 && l1 < LL) ? 1.0f : 0.0f;
      v2f a;
      a.x = hRow[(size_t)i0 * LL + l0c] * m0;
      a.y = hRow[(size_t)i0 * LL + l1c] * m1;
      // B from LDS: lds[(i0-s*32)*512 + o*8 + q], rows kk0/kk0+1 adjacent
      const float* bp = ldsW + (size_t)(i0 - s * 32) * 512 + colL * 8 + q0;
      v2f b0; b0.x = bp[0];   b0.y = bp[1];
      v2f b1; b1.x = bp[128]; b1.y = bp[129];
      v2f b2; b2.x = bp[256]; b2.y = bp[257];
      v2f b3; b3.x = bp[384]; b3.y = bp[385];
      a0 = __builtin_amdgcn_wmma_f32_16x16x4_f32(false, a, false, b0, (short)0, a0, false, false);
      a1 = __builtin_amdgcn_wmma_f32_16x16x4_f32(false, a, false, b1, (short)0, a1, false, false);
      a2 = __builtin_amdgcn_wmma_f32_16x16x4_f32(false, a, false, b2, (short)0, a2, false, false);
      a3 = __builtin_amdgcn_wmma_f32_16x16x4_f32(false, a, false, b3, (short)0, a3, false, false);
    }
  }

  // ---- epilogue: bias + residual, scatter to (B,C,T,F) ----
  const int rOff = (lane >> 4) << 3;
#pragma unroll
  for (int v = 0; v < 8; ++v) {
    int r = tileM * 16 + v + rOff;
    int nr = r / TT, tr = r % TT;
    int b_ = nr / FF, f_ = nr % FF;
#pragma unroll
    for (int j = 0; j < 4; ++j) {
      int o = j * 16 + colL;
      float av = (j == 0) ? a0[v] : (j == 1) ? a1[v] : (j == 2) ? a2[v] : a3[v];
      size_t oidx = (((size_t)b_ * CC + o) * TT + tr) * FF + f_;
      out[oidx] = av + bct[o] + x[oidx];        // residual (no padding case)
    }
  }
}

// =====================================================================
extern "C" void kernel_launch(void* const* d_in, const int* in_sizes, int n_in,
                              void* d_out, int out_size, void* d_ws, size_t ws_size,
                              hipStream_t stream) {
  const float* x     = (const float*)d_in[0];
  const float* gamma = (const float*)d_in[1];
  const float* beta  = (const float*)d_in[2];
  const float* W1f   = (const float*)d_in[3];
  const float* b1f   = (const float*)d_in[4];
  const float* W1b   = (const float*)d_in[5];
  const float* b1b   = (const float*)d_in[6];
  const float* W2f   = (const float*)d_in[7];
  const float* b2f   = (const float*)d_in[8];
  const float* W2b   = (const float*)d_in[9];
  const float* b2b   = (const float*)d_in[10];
  const float* Wct   = (const float*)d_in[11];
  const float* bct   = (const float*)d_in[12];

  float* ws = (float*)d_ws;
  float* xr = ws + WS_XR;
  float* U  = ws + WS_U;
  float* H  = ws + WS_H;
  float* out = (float*)d_out;

  // 1) layernorm -> xr
  {
    int nthr = BB * TT * FF;
    k_layernorm<<<(nthr + 255) / 256, 256, 0, stream>>>(x, gamma, beta, xr);
  }
  // 2) SRU projections: (M1 x 256 x 256) per group, fp32 WMMA + TDM/LDS
  {
    int mTiles = (M1 + 15) / 16;                  // 8031
    dim3 grid((mTiles + 7) / 8, 4, 2);            // (1004, 4, 2)
    k_sru_gemm<<<grid, 256, KPG * 64 * sizeof(float), stream>>>(
        xr, W1f, W1b, W2f, W2b, U);
  }
  // 3) scans (4 dirs x 516 x 32 lanes)
  {
    int nthr = 4 * NN * HG;
    k_sru_scan<<<(nthr + 255) / 256, 256, 0, stream>>>(U, b1f, b1b, b2f, b2b, H);
  }
  // 4) conv-transpose GEMM + bias + residual -> d_out
  {
    int mTiles = M2 / 16;                         // 8256
    k_convt<<<dim3(mTiles / 8), 256, 32 * 512 * sizeof(float), stream>>>(
        H, Wct, bct, x, out);
  }
}